// EmbeddingLayer_13975823582157
// MI455X (gfx1250) — compile-verified
//
#include <hip/hip_runtime.h>
#include <hip/hip_bf16.h>

// ---------------------------------------------------------------------------
// EmbeddingLayer for MI455X (gfx1250, wave32, WMMA).
//
// Fusion: char_out = avg_ce @ (lin_w @ conv_w)^T + (lin_w @ conv_b + lin_b)
// (conv(k=1) and masked-mean are linear, so they commute; the two linears
// collapse into one 100x64 fused weight). The char path is then a single
// [16384 x 64] x [64 x 100] fp32 GEMM done with V_WMMA_F32_16X16X4_F32.
// Problem is memory bound (~47 MB traffic, ~0.2 GFLOP), so everything else
// is a streaming gather.
// ---------------------------------------------------------------------------

typedef __attribute__((ext_vector_type(2))) float v2f;
typedef __attribute__((ext_vector_type(8))) float v8f;

#define N_ROWS   (32 * 512)   // B*S
#define L_CHARS  16
#define CHAR_EMB 64
#define OUT_DIM  100
#define OUT_PAD  112          // 7 tiles of 16 cols
#define GLOVE_D  300
#define OUT_W    400          // GLOVE_D + OUT_DIM
#define TAB_ROWS 128
#define LDS_STRIDE 68         // 64 + 4 pad: stride%64==4 -> conflict-free B reads

// --- Kernel 1: fuse (lin_w @ conv_w) -> fW[112][64], fused bias -> fb[112] ---
__global__ __launch_bounds__(256) void fuse_weights(
    const float* __restrict__ conv_w, const float* __restrict__ conv_b,
    const float* __restrict__ lin_w,  const float* __restrict__ lin_b,
    float* __restrict__ fW, float* __restrict__ fb) {
  int t = blockIdx.x * 256 + threadIdx.x;
  if (t >= OUT_PAD * CHAR_EMB) return;
  int o = t >> 6, e = t & 63;
  float s = 0.f;
  if (o < OUT_DIM) {
    for (int j = 0; j < OUT_DIM; ++j)
      s += lin_w[o * OUT_DIM + j] * conv_w[j * CHAR_EMB + e];
  }
  fW[t] = s;
  if (e == 0) {
    float b = 0.f;
    if (o < OUT_DIM) {
      for (int j = 0; j < OUT_DIM; ++j) b += lin_w[o * OUT_DIM + j] * conv_b[j];
      b += lin_b[o];
    }
    fb[o] = b;
  }
}

// --- Kernel 2: GloVe gather, one wave (32 lanes) per row, float4 streams ----
__global__ __launch_bounds__(256) void glove_gather(
    const int* __restrict__ tokens, const float* __restrict__ glove,
    float* __restrict__ out) {
  int wave = threadIdx.x >> 5;
  int lane = threadIdx.x & 31;
  int row  = blockIdx.x * 8 + wave;
  int tok  = tokens[row];
  const float4* src = (const float4*)(glove + (size_t)tok * GLOVE_D);  // 75 f4
  float4*       dst = (float4*)(out + (size_t)row * OUT_W);
  #pragma unroll
  for (int i = lane; i < GLOVE_D / 4; i += 32) dst[i] = src[i];
}

// --- Kernel 3: char path. 8 waves/block, 16 rows/wave, WMMA f32 16x16x4 -----
__global__ __launch_bounds__(256) void char_path(
    const int* __restrict__ char_ids, const int* __restrict__ char_lens,
    const float* __restrict__ char_table,
    const float* __restrict__ fW, const float* __restrict__ fb,
    float* __restrict__ out) {
  __shared__ float s_tab[TAB_ROWS * LDS_STRIDE];   // 34.0 KB char embedding table
  __shared__ float s_W[OUT_PAD * LDS_STRIDE];      // 29.8 KB fused weights
  __shared__ float s_b[OUT_PAD];
  __shared__ int   s_ids[8][256];                  // per-wave 16x16 char ids
  __shared__ int   s_len[8][16];

  const int tid  = threadIdx.x;
  const int wave = tid >> 5;
  const int lane = tid & 31;
  const int r    = lane & 15;   // row-in-tile (A) == col-in-tile (B/D N index)
  const int half = lane >> 4;   // K-half selector per ISA A/B f32 layouts
  const int rowBase = (blockIdx.x * 8 + wave) * 16;

  // Stage char_table and fused W into LDS (strided to kill bank conflicts).
  const float4* t4 = (const float4*)char_table;
  for (int i4 = tid; i4 < TAB_ROWS * CHAR_EMB / 4; i4 += 256) {
    float4 v = t4[i4];
    int i = i4 * 4, o = i >> 6, e = i & 63;
    *(float4*)&s_tab[o * LDS_STRIDE + e] = v;
  }
  const float4* w4 = (const float4*)fW;
  for (int i4 = tid; i4 < OUT_PAD * CHAR_EMB / 4; i4 += 256) {
    float4 v = w4[i4];
    int i = i4 * 4, o = i >> 6, e = i & 63;
    *(float4*)&s_W[o * LDS_STRIDE + e] = v;
  }
  if (tid < OUT_PAD) s_b[tid] = fb[tid];
  // Stage this wave's 16x16 char-id tile (256 ints = 2 int4 per lane).
  {
    const int4* ids4 = (const int4*)(char_ids + (size_t)rowBase * L_CHARS);
    int4* sd = (int4*)&s_ids[wave][0];
    sd[lane]      = ids4[lane];
    sd[lane + 32] = ids4[lane + 32];
    if (lane < 16) s_len[wave][lane] = char_lens[rowBase + lane];
  }
  __syncthreads();

  // Accumulate avg_ce directly in the WMMA A-fragment register layout:
  // lane owns row r, columns 4*p + 2*half + {0,1} for p = 0..15.
  const int len = s_len[wave][r];
  float acc[32];
  #pragma unroll
  for (int j = 0; j < 32; ++j) acc[j] = 0.f;
  for (int c = 0; c < L_CHARS; ++c) {
    if (c < len) {
      int id = s_ids[wave][r * L_CHARS + c];
      const float* trow = &s_tab[id * LDS_STRIDE + 2 * half];
      #pragma unroll
      for (int p = 0; p < 16; ++p) {
        v2f v = *(const v2f*)&trow[4 * p];
        acc[2 * p]     += v.x;
        acc[2 * p + 1] += v.y;
      }
    }
  }
  const float inv = 1.0f / (float)len;
  #pragma unroll
  for (int j = 0; j < 32; ++j) acc[j] *= inv;

  // GEMM: D[16x16] += A[16x4] x B[4x16] over 16 k-steps, 7 column tiles.
  for (int ct = 0; ct < 7; ++ct) {
    v8f cacc = {};
    const float* wb = &s_W[(ct * 16 + r) * LDS_STRIDE + 2 * half];
    #pragma unroll
    for (int kk = 0; kk < 16; ++kk) {
      v2f a = { acc[2 * kk], acc[2 * kk + 1] };
      v2f b = *(const v2f*)&wb[4 * kk];
      cacc = __builtin_amdgcn_wmma_f32_16x16x4_f32(
          /*neg_a=*/false, a, /*neg_b=*/false, b,
          /*c_mod=*/(short)0, cacc, /*reuse_a=*/false, /*reuse_b=*/false);
    }
    int col = ct * 16 + r;            // D: N = lane%16
    if (col < OUT_DIM) {
      float bias = s_b[col];
      #pragma unroll
      for (int v = 0; v < 8; ++v) {   // D: M = v + 8*(lane/16)
        int row = rowBase + v + 8 * half;
        out[(size_t)row * OUT_W + GLOVE_D + col] = cacc[v] + bias;
      }
    }
  }
}

extern "C" void kernel_launch(void* const* d_in, const int* in_sizes, int n_in,
                              void* d_out, int out_size, void* d_ws, size_t ws_size,
                              hipStream_t stream) {
  const int*   word_tokens = (const int*)d_in[0];
  const int*   char_ids    = (const int*)d_in[1];
  const int*   char_lens   = (const int*)d_in[2];
  const float* glove_table = (const float*)d_in[3];
  const float* char_table  = (const float*)d_in[4];
  const float* conv_w      = (const float*)d_in[5];
  const float* conv_b      = (const float*)d_in[6];
  const float* lin_w       = (const float*)d_in[7];
  const float* lin_b       = (const float*)d_in[8];
  float* out = (float*)d_out;

  float* fW = (float*)d_ws;            // 112*64 floats
  float* fb = fW + OUT_PAD * CHAR_EMB; // 112 floats

  fuse_weights<<<(OUT_PAD * CHAR_EMB + 255) / 256, 256, 0, stream>>>(
      conv_w, conv_b, lin_w, lin_b, fW, fb);
  glove_gather<<<N_ROWS / 8, 256, 0, stream>>>(word_tokens, glove_table, out);
  char_path<<<N_ROWS / 128, 256, 0, stream>>>(char_ids, char_lens, char_table,
                                              fW, fb, out);
}